// AFM_PRBS_1597727834141
// MI455X (gfx1250) — compile-verified
//
#include <hip/hip_runtime.h>
#include <math.h>

typedef float v2f __attribute__((ext_vector_type(2)));
typedef float v8f __attribute__((ext_vector_type(8)));

#define B_   16
#define H_   256
#define W_   256
#define HW_  (H_ * W_)

// ---------------------------------------------------------------------------
// 1) filter_input = [ |y|, log10(|y|+1e-10), |z|, log10(|z|+1e-10) ]
// ---------------------------------------------------------------------------
__global__ void prep_kernel(const float* __restrict__ z,
                            const float* __restrict__ y,
                            float* __restrict__ filt) {
    int t = blockIdx.x * blockDim.x + threadIdx.x;   // B*HW threads
    if (t >= B_ * HW_) return;
    int b = t / HW_;
    int p = t - b * HW_;
    size_t base = (size_t)b * 2 * HW_ + p;
    float yr = y[base], yi = y[base + HW_];
    float zr = z[base], zi = z[base + HW_];
    float ay = sqrtf(yr * yr + yi * yi);
    float az = sqrtf(zr * zr + zi * zi);
    float* fb = filt + (size_t)b * 4 * HW_;
    fb[p]            = ay;
    fb[HW_ + p]      = log10f(ay + 1e-10f);
    fb[2 * HW_ + p]  = az;
    fb[3 * HW_ + p]  = log10f(az + 1e-10f);
}

// ---------------------------------------------------------------------------
// Implicit-GEMM 3x3 SAME conv inner loop on V_WMMA_F32_16X16X4_F32.
//   A (16x4)  : weights, (OC, IC*9) row-major == flattened OIHW.
//               lanes 0-15 -> K = k0+0 (v0), k0+1 (v1); lanes 16-31 -> +2,+3
//   B (4x16)  : im2col column for pixel n = lane&15 (rows h and h+1 share A)
//   C (16x16) : VGPR r -> oc_local = r + 8*(lane>>4), pixel = lane&15
// Fully unrolled so each step's (ic,kh,kw) pair folds to compile-time
// constants; the only lane-dependence is a 2-way select (lane<16 vs >=16).
// Boundary handling is branchless: offset clamped in-tensor, value masked.
// ---------------------------------------------------------------------------
template<int IC, int INH, int INW, bool TWO_ROWS>
__device__ __forceinline__ void conv_rows_wmma(
    const float* __restrict__ inb,   // (IC, INH, INW) for this batch
    const float* __restrict__ wts,   // (OC, IC*9) row-major
    int oc0, int h, int w0, int lane,
    v8f& c0, v8f& c1)
{
    constexpr int K   = IC * 9;
    constexpr int CHW = IC * INH * INW;
    const int  mn = lane & 15;
    const bool hi = (lane & 16) != 0;
    const float* arow = wts + (size_t)(oc0 + mn) * K + (hi ? 2 : 0);
    const int wq = w0 + mn;

    // loop-invariant validity of the 3 kernel offsets per lane
    bool vh0[3], vh1[3], vw[3];
    #pragma unroll
    for (int t = 0; t < 3; ++t) {
        vh0[t] = (unsigned)(h + t - 1) < (unsigned)INH;   // row h   : hh = h+t-1
        vh1[t] = (unsigned)(h + t)     < (unsigned)INH;   // row h+1 : hh = h+t
        vw[t]  = (unsigned)(wq + t - 1) < (unsigned)INW;
    }
    const int pbase = h * INW + wq;   // pixel base (row h, col wq)

    #pragma unroll
    for (int k0 = 0; k0 < K; k0 += 4) {
        v2f a;
        a.x = arow[k0];
        a.y = arow[k0 + 1];

        float b0v[2], b1v[2];
        #pragma unroll
        for (int u = 0; u < 2; ++u) {
            // lane<16 uses k = k0+u ("A"), lane>=16 uses k = k0+u+2 ("B");
            // both decompositions are compile-time constants after unroll.
            int kA = k0 + u;
            int icA = kA / 9;  int rA = kA - icA * 9;
            int khA = rA / 3;  int kwA = rA - 3 * khA;
            int kB = k0 + u + 2;
            int icB = kB / 9;  int rB = kB - icB * 9;
            int khB = rB / 3;  int kwB = rB - 3 * khB;
            const int offA = icA * INH * INW + (khA - 1) * INW + (kwA - 1);
            const int offB = icB * INH * INW + (khB - 1) * INW + (kwB - 1);

            bool vA0 = vh0[khA] && vw[kwA];
            bool vB0 = vh0[khB] && vw[kwB];
            int  off    = hi ? offB : offA;
            bool valid0 = hi ? vB0  : vA0;

            int lin  = pbase + off;
            int linc = lin < 0 ? 0 : (lin > CHW - 1 ? CHW - 1 : lin);
            float r0 = inb[linc];
            b0v[u]   = valid0 ? r0 : 0.0f;

            if (TWO_ROWS) {
                bool vA1 = vh1[khA] && vw[kwA];
                bool vB1 = vh1[khB] && vw[kwB];
                bool valid1 = hi ? vB1 : vA1;
                int lin1  = lin + INW;
                int linc1 = lin1 < 0 ? 0 : (lin1 > CHW - 1 ? CHW - 1 : lin1);
                float r1  = inb[linc1];
                b1v[u]    = valid1 ? r1 : 0.0f;
            }
        }
        v2f b0; b0.x = b0v[0]; b0.y = b0v[1];
        c0 = __builtin_amdgcn_wmma_f32_16x16x4_f32(false, a, false, b0,
                                                   (short)0, c0, false, false);
        if (TWO_ROWS) {
            v2f b1; b1.x = b1v[0]; b1.y = b1v[1];
            c1 = __builtin_amdgcn_wmma_f32_16x16x4_f32(false, a, false, b1,
                                                       (short)0, c1, false, false);
        }
    }
}

// ---------------------------------------------------------------------------
// 2/3) WMMA conv + ReLU + 2x2 avgpool: (B,IC,INH,INW) -> (B,OC,INH/2,INW/2)
//      One wave per (b, oc-tile, pooled row, 8-pooled-col group).
// ---------------------------------------------------------------------------
template<int IC, int OC, int INH, int INW>
__global__ void __launch_bounds__(256)
conv_pool_wmma_kernel(const float* __restrict__ in0,
                      const float* __restrict__ wts,
                      const float* __restrict__ bias,
                      float* __restrict__ out) {
    __shared__ float sm[8][2][16][16];
    const int lane = threadIdx.x & 31;
    const int wv   = threadIdx.x >> 5;
    int wid = blockIdx.x * 8 + wv;
    constexpr int NWG = INW / 16;
    constexpr int NPH = INH / 2;
    constexpr int NOT = OC / 16;
    int pwg = wid % NWG;  wid /= NWG;
    int ph  = wid % NPH;  wid /= NPH;
    int oct = wid % NOT;
    int b   = wid / NOT;
    const int oc0 = oct * 16;
    const int h   = 2 * ph;
    const int w0  = pwg * 16;
    const float* inb = in0 + (size_t)b * IC * INH * INW;

    v8f c0 = {}, c1 = {};
    conv_rows_wmma<IC, INH, INW, true>(inb, wts, oc0, h, w0, lane, c0, c1);

    const int mh = (lane >> 4) * 8;
    const int n  = lane & 15;
    #pragma unroll
    for (int r = 0; r < 8; ++r) {
        float bv = bias[oc0 + r + mh];
        sm[wv][0][r + mh][n] = fmaxf(c0[r] + bv, 0.0f);
        sm[wv][1][r + mh][n] = fmaxf(c1[r] + bv, 0.0f);
    }
    __syncthreads();

    constexpr int POW = INW / 2;
    #pragma unroll
    for (int j = 0; j < 4; ++j) {
        int o  = lane + 32 * j;          // 0..127 : 16 oc x 8 pooled cols
        int oc = o >> 3;
        int pj = o & 7;
        float v = 0.25f * (sm[wv][0][oc][2 * pj] + sm[wv][0][oc][2 * pj + 1] +
                           sm[wv][1][oc][2 * pj] + sm[wv][1][oc][2 * pj + 1]);
        out[(((size_t)b * OC + oc0 + oc) * NPH + ph) * POW + pwg * 8 + pj] = v;
    }
}

// ---------------------------------------------------------------------------
// 4a) WMMA conv3 + ReLU + per-wave column reduction -> partial sums
//     (B,32,64,64) -> part[(b*64+oc)][256]  (256 = 64 rows x 4 pixel groups)
// ---------------------------------------------------------------------------
__global__ void __launch_bounds__(256)
conv3_part_wmma_kernel(const float* __restrict__ in0,
                       const float* __restrict__ wts,   // (64, 288)
                       const float* __restrict__ bias,
                       float* __restrict__ part) {
    __shared__ float sm[8][16][16];
    const int lane = threadIdx.x & 31;
    const int wv   = threadIdx.x >> 5;
    int wid = blockIdx.x * 8 + wv;       // 16384 waves total
    int pg  = wid & 3;
    int h   = (wid >> 2) & 63;
    int oct = (wid >> 8) & 3;
    int b   = wid >> 10;
    const int oc0 = oct * 16;
    const int w0  = pg * 16;
    const float* inb = in0 + (size_t)b * 32 * 64 * 64;

    v8f c0 = {}, cunused = {};
    conv_rows_wmma<32, 64, 64, false>(inb, wts, oc0, h, w0, lane, c0, cunused);

    const int mh = (lane >> 4) * 8;
    const int n  = lane & 15;
    #pragma unroll
    for (int r = 0; r < 8; ++r)
        sm[wv][r + mh][n] = fmaxf(c0[r] + bias[oc0 + r + mh], 0.0f);
    __syncthreads();

    if (lane < 16) {
        float s = 0.0f;
        #pragma unroll
        for (int q = 0; q < 16; ++q) s += sm[wv][lane][q];
        part[((size_t)b * 64 + oc0 + lane) * 256 + (h * 4 + pg)] = s;
    }
}

// 4b) deterministic partial reduction -> feat (B,64), scaled by 1/4096
__global__ void feat_reduce_kernel(const float* __restrict__ part,
                                   float* __restrict__ feat) {
    int i = blockIdx.x * blockDim.x + threadIdx.x;   // 1024 = B*64
    if (i >= B_ * 64) return;
    const float* p = part + (size_t)i * 256;
    float s = 0.0f;
    for (int j = 0; j < 256; ++j) s += p[j];
    feat[i] = s * (1.0f / 4096.0f);
}

// ---------------------------------------------------------------------------
// 5) FP32 WMMA GEMM: out(16 x N) = A(16 x K) * W^T + bias, W is (N x K) rowmaj
// ---------------------------------------------------------------------------
__global__ void __launch_bounds__(32)
gemm_wmma_f32(const float* __restrict__ A,
              const float* __restrict__ W,
              const float* __restrict__ bias,
              float* __restrict__ out,
              int K, int N) {
    const int lane  = threadIdx.x;
    const int n0    = blockIdx.x * 16;
    const int mn    = lane & 15;
    const int khalf = (lane >> 4) << 1;
    v8f c = {};
    const float* arow = A + (size_t)mn * K + khalf;
    const float* wrow = W + (size_t)(n0 + mn) * K + khalf;
    for (int k0 = 0; k0 < K; k0 += 4) {
        __builtin_prefetch((const void*)(wrow + k0 + 64), 0, 1);
        v2f a, b;
        a.x = arow[k0];
        a.y = arow[k0 + 1];
        b.x = wrow[k0];
        b.y = wrow[k0 + 1];
        c = __builtin_amdgcn_wmma_f32_16x16x4_f32(false, a, false, b,
                                                  (short)0, c, false, false);
    }
    const float bn = bias[n0 + mn];
    const int mh = (lane >> 4) * 8;
    #pragma unroll
    for (int r = 0; r < 8; ++r) {
        out[(size_t)(r + mh) * N + (n0 + mn)] = c[r] + bn;
    }
}

// ---------------------------------------------------------------------------
// 6) per (b,row): value_set = sum_j softmax(logits*0.1)[j] * 0.01*(j+1)
// ---------------------------------------------------------------------------
__global__ void softmax_value_kernel(const float* __restrict__ logits,
                                     float* __restrict__ vset) {
    int t = blockIdx.x * blockDim.x + threadIdx.x;
    if (t >= B_ * 100) return;
    int b   = t / 100;
    int row = t - b * 100;
    const float* x = logits + (size_t)b * 10000 + row * 100;
    float m = -INFINITY;
    for (int j = 0; j < 100; ++j) m = fmaxf(m, x[j]);
    float s = 0.0f, wsum = 0.0f;
    for (int j = 0; j < 100; ++j) {
        float e = expf(0.1f * (x[j] - m));
        s    += e;
        wsum += e * (0.01f * (float)(j + 1));
    }
    vset[t] = wsum / s;   // * FQ_BOUND(=1)
}

// ---------------------------------------------------------------------------
// 7) radial gather + complex replacement; writes all three outputs
// ---------------------------------------------------------------------------
__global__ void finalize_kernel(const float* __restrict__ z,
                                const float* __restrict__ y,
                                const float* __restrict__ prbs,
                                const float* __restrict__ vset,
                                float* __restrict__ out) {
    int t = blockIdx.x * blockDim.x + threadIdx.x;   // B*HW
    if (t >= B_ * HW_) return;
    int b = t / HW_;
    int p = t - b * HW_;
    int h  = p >> 8;
    int wq = p & 255;

    const float max_radius = 181.0193359837562f;     // sqrt(256^2+256^2)/2
    float dy = (float)h  - 128.0f;
    float dx = (float)wq - 128.0f;
    float dist = sqrtf(dy * dy + dx * dx);

    // searchsorted(radius_set, dist, 'right'); radius_set[k] = max_radius*0.01*(k+1)
    int idx = (int)floorf(dist / (max_radius * 0.01f));
    if (idx < 0) idx = 0;
    if (idx > 100) idx = 100;
    while (idx > 0   && max_radius * (0.01f * (float)idx)       >  dist) --idx;
    while (idx < 100 && max_radius * (0.01f * (float)(idx + 1)) <= dist) ++idx;

    int ic = idx < 99 ? idx : 99;
    float gathered = vset[b * 100 + ic];
    float mask = (idx < 100) ? gathered : 0.0f;

    size_t base = (size_t)b * 2 * HW_ + p;
    float zr = z[base],    zi = z[base + HW_];
    float yr = y[base],    yi = y[base + HW_];
    float pr = prbs[base], pi = prbs[base + HW_];

    float mag2 = fmaxf(pr * pr + pi * pi, 1e-10f);
    float inv  = 1.0f / mag2;

    // replaced_z = z + (pr,-pi) * (y - z*mask)
    float t1r = yr - zr * mask, t1i = yi - zi * mask;
    float rz_r = zr + (pr * t1r + pi * t1i);
    float rz_i = zi + (pr * t1i - pi * t1r);

    // replaced_y = y + (pr,-pi)/mag2 * (z - y*mask)
    float t2r = zr - yr * mask, t2i = zi - yi * mask;
    float scr = pr * inv, sci = -pi * inv;
    float ry_r = yr + (scr * t2r - sci * t2i);
    float ry_i = yi + (scr * t2i + sci * t2r);

    float* out_z = out;
    float* out_y = out + (size_t)B_ * 2 * HW_;
    float* out_m = out + (size_t)B_ * 4 * HW_;
    out_z[base]       = rz_r;
    out_z[base + HW_] = rz_i;
    out_y[base]       = ry_r;
    out_y[base + HW_] = ry_i;
    out_m[(size_t)b * HW_ + p] = mask;
}

// ---------------------------------------------------------------------------
extern "C" void kernel_launch(void* const* d_in, const int* in_sizes, int n_in,
                              void* d_out, int out_size, void* d_ws, size_t ws_size,
                              hipStream_t stream) {
    const float* z    = (const float*)d_in[0];
    const float* y    = (const float*)d_in[1];
    const float* prbs = (const float*)d_in[2];
    const float* c1w  = (const float*)d_in[3];
    const float* c1b  = (const float*)d_in[4];
    const float* c2w  = (const float*)d_in[5];
    const float* c2b  = (const float*)d_in[6];
    const float* c3w  = (const float*)d_in[7];
    const float* c3b  = (const float*)d_in[8];
    const float* f1w  = (const float*)d_in[9];
    const float* f1b  = (const float*)d_in[10];
    const float* f2w  = (const float*)d_in[11];
    const float* f2b  = (const float*)d_in[12];

    float* wsf    = (float*)d_ws;
    float* filt   = wsf;                       // B*4*HW       = 4,194,304
    float* p1     = filt   + 4194304;          // B*16*128*128 = 4,194,304
    float* p2     = p1     + 4194304;          // B*32*64*64   = 2,097,152
    float* part   = p2     + 2097152;          // B*64*256     =   262,144
    float* feat   = part   + 262144;           // B*64         =     1,024
    float* v1     = feat   + 1024;             // B*256        =     4,096
    float* logits = v1     + 4096;             // B*10000      =   160,000
    float* vset   = logits + 160000;           // B*100        =     1,600
    float* out    = (float*)d_out;

    prep_kernel<<<4096, 256, 0, stream>>>(z, y, filt);

    // conv1: waves = 16 * 1 * 128 * 16 = 32768 -> 4096 blocks of 8 waves
    conv_pool_wmma_kernel<4, 16, 256, 256><<<4096, 256, 0, stream>>>(filt, c1w, c1b, p1);
    // conv2: waves = 16 * 2 * 64 * 8 = 16384 -> 2048 blocks
    conv_pool_wmma_kernel<16, 32, 128, 128><<<2048, 256, 0, stream>>>(p1, c2w, c2b, p2);
    // conv3: waves = 16 * 4 * 64 * 4 = 16384 -> 2048 blocks
    conv3_part_wmma_kernel<<<2048, 256, 0, stream>>>(p2, c3w, c3b, part);
    feat_reduce_kernel<<<4, 256, 0, stream>>>(part, feat);

    gemm_wmma_f32<<<16,  32, 0, stream>>>(feat, f1w, f1b, v1, 64, 256);
    gemm_wmma_f32<<<625, 32, 0, stream>>>(v1, f2w, f2b, logits, 256, 10000);

    softmax_value_kernel<<<7, 256, 0, stream>>>(logits, vset);
    finalize_kernel<<<4096, 256, 0, stream>>>(z, y, prbs, vset, out);
}